// H_H_EdgeApplyMoudle_712964571353
// MI455X (gfx1250) — compile-verified
//
#include <hip/hip_runtime.h>
#include <hip/hip_bf16.h>

typedef __attribute__((ext_vector_type(2))) float v2f;
typedef __attribute__((ext_vector_type(4))) float v4f;
typedef __attribute__((ext_vector_type(8))) float v8f;

constexpr int D_NODE   = 128;   // node feature width
constexpr int IN_FEAT  = 256;   // 2 * D_NODE
constexpr int OUT_FEAT = 128;
constexpr int M_BLOCK  = 128;   // edges per workgroup

// LDS: swizzled feat tile [128 rows x 256 f32] then swizzled W [128 x 256 f32]
constexpr int FEAT_BYTES = M_BLOCK  * IN_FEAT * 4;   // 128 KB
constexpr int W_BYTES    = OUT_FEAT * IN_FEAT * 4;   // 128 KB
constexpr int LDS_BYTES  = FEAT_BYTES + W_BYTES;     // 256 KB (<= 320 KB/WGP)

// Row layout: 256 f32 = 128 8-byte "pairs"; physical pair = p ^ (2*(row&15)).
// Fragment read (A and B identical pattern, per ISA 16x16x4 layout):
//   lane L -> row r0+(L&15), pair 2t+(L>>4)  => 64 distinct LDS dword banks.
__device__ __forceinline__ v2f ld_frag(const char* base, int r0, int t, int lane) {
    const int row  = r0 + (lane & 15);
    const int p    = (t << 1) + (lane >> 4);
    const int phys = p ^ ((lane & 15) << 1);
    return *(const v2f*)(base + row * 1024 + phys * 8);
}

__global__ void __launch_bounds__(256)
gat_edge_apply_kernel(const float* __restrict__ nf,      // [50000,128]
                      const float* __restrict__ wEdge,   // [128,256]
                      const int*   __restrict__ srcIdx,  // [E]
                      const int*   __restrict__ dstIdx,  // [E]
                      float* __restrict__ eFeat,         // [E,128]
                      float* __restrict__ alpha,         // [E]
                      int nEdges)
{
    extern __shared__ char smem[];
    char* featS = smem;
    char* wS    = smem + FEAT_BYTES;

    const int tid   = threadIdx.x;
    const int lane  = tid & 31;
    const int wave  = tid >> 5;
    const int eBase = blockIdx.x * M_BLOCK;

    // softmax over a singleton axis == 1.0 (W_attn / leaky-relu are dead code)
    if (tid < M_BLOCK) {
        const int e = eBase + tid;
        if (e < nEdges) alpha[e] = 1.0f;
    }

    // ---------------- LDS fill (coalesced 512B bursts, swizzled stores) -----
    // 256 (row,half) fills for feat + 256 for W; each wave does 32 of each.
    #pragma unroll 4
    for (int i = 0; i < 32; ++i) {
        const int rr  = wave * 32 + i;   // 0..255
        const int row = rr >> 1;         // edge-in-block / out-feature row
        const int h   = rr & 1;          // which 128-float half of the 256 row
        const int q   = ((h << 6) + (lane << 1)) ^ ((row & 15) << 1);

        // gathered edge features: first half = src node, second half = dst node
        {
            int eg = eBase + row;
            if (eg >= nEdges) eg = nEdges - 1;      // clamp (stores are predicated)
            const long node = (h ? dstIdx : srcIdx)[eg];
            const v4f v = *(const v4f*)(nf + node * (long)D_NODE + lane * 4);
            *(v4f*)(featS + row * 1024 + q * 8) = v;
        }
        // W_edge tile (whole matrix, reused by all 8 waves)
        {
            const v4f v = *(const v4f*)(wEdge + (size_t)row * IN_FEAT + h * 128 + lane * 4);
            *(v4f*)(wS + row * 1024 + q * 8) = v;
        }
    }
    __syncthreads();

    // ---------------- WMMA main loop ---------------------------------------
    // wave -> 32 edges (2 M-tiles) x 64 out-features (4 N-tiles), K = 256
    const int mg = wave >> 1;        // 0..3
    const int ng = wave & 1;         // 0..1
    const int e0 = mg * 32;
    const int n0 = ng * 64;

    v8f acc[2][4];
    #pragma unroll
    for (int mi = 0; mi < 2; ++mi)
        #pragma unroll
        for (int nj = 0; nj < 4; ++nj)
            acc[mi][nj] = (v8f){0.f,0.f,0.f,0.f,0.f,0.f,0.f,0.f};

    for (int t = 0; t < 64; ++t) {   // K/4 steps
        const v2f a0 = ld_frag(featS, e0,      t, lane);
        const v2f a1 = ld_frag(featS, e0 + 16, t, lane);
        #pragma unroll
        for (int nj = 0; nj < 4; ++nj) {
            const v2f b = ld_frag(wS, n0 + nj * 16, t, lane);
            acc[0][nj] = __builtin_amdgcn_wmma_f32_16x16x4_f32(
                false, a0, false, b, (short)0, acc[0][nj], false, false);
            acc[1][nj] = __builtin_amdgcn_wmma_f32_16x16x4_f32(
                false, a1, false, b, (short)0, acc[1][nj], false, false);
        }
    }

    // ---------------- sigmoid epilogue + coalesced stores -------------------
    // D layout: VGPR r, lanes 0-15 -> (M=r, N=lane), lanes 16-31 -> (M=r+8).
    const int ml = lane & 15;
    const int hb = lane >> 4;
    #pragma unroll
    for (int mi = 0; mi < 2; ++mi) {
        const int erow = eBase + e0 + mi * 16 + hb * 8;
        #pragma unroll
        for (int nj = 0; nj < 4; ++nj) {
            const int col = n0 + nj * 16 + ml;
            #pragma unroll
            for (int r = 0; r < 8; ++r) {
                const int e = erow + r;
                if (e < nEdges) {
                    const float x = acc[mi][nj][r];
                    eFeat[(size_t)e * OUT_FEAT + col] = 1.0f / (1.0f + __expf(-x));
                }
            }
        }
    }
}

extern "C" void kernel_launch(void* const* d_in, const int* in_sizes, int n_in,
                              void* d_out, int out_size, void* d_ws, size_t ws_size,
                              hipStream_t stream) {
    const float* nf     = (const float*)d_in[0];   // n_f    [50000*128]
    const float* wEdge  = (const float*)d_in[1];   // W_edge [128*256]
    // d_in[2] = W_attn — provably unused (softmax over singleton axis == 1)
    const int*   srcIdx = (const int*)d_in[3];
    const int*   dstIdx = (const int*)d_in[4];
    const int nEdges = in_sizes[3];

    float* eFeat = (float*)d_out;
    float* alpha = (float*)d_out + (size_t)nEdges * OUT_FEAT;

    (void)d_ws; (void)ws_size; (void)n_in; (void)out_size;

    // 256 KB dynamic LDS > default 64 KB cap: raise the attribute (deterministic).
    hipFuncSetAttribute((const void*)gat_edge_apply_kernel,
                        hipFuncAttributeMaxDynamicSharedMemorySize, LDS_BYTES);

    const int grid = (nEdges + M_BLOCK - 1) / M_BLOCK;   // 4688 blocks
    gat_edge_apply_kernel<<<grid, 256, LDS_BYTES, stream>>>(
        nf, wEdge, srcIdx, dstIdx, eFeat, alpha, nEdges);
}